// Controller_17008070492262
// MI455X (gfx1250) — compile-verified
//
#include <hip/hip_runtime.h>
#include <hip/hip_bf16.h>
#include <math.h>

#define HIDDEN 2048
#define CHOICE_NUM 19

typedef __attribute__((ext_vector_type(2))) float v2f;
typedef __attribute__((ext_vector_type(8))) float v8f;

__device__ __forceinline__ float sigmoidf_(float v) {
    return 1.0f / (1.0f + expf(-v));
}

// Gather the selected embedding row into the workspace x vector.
__global__ void embed_gather_kernel(const int* __restrict__ idx,
                                    const float* __restrict__ emb,
                                    float* __restrict__ x) {
    int i = blockIdx.x * blockDim.x + threadIdx.x;
    if (i < HIDDEN) x[i] = emb[(size_t)idx[0] * HIDDEN + i];
}

// One LSTM layer with zero initial state.
//   gates = W_ih x + b_ih + b_hh ; h = sigmoid(o) * tanh(sigmoid(i) * tanh(g))
// f-gate and all of w_hh are skipped (they multiply zeros).
// Grid: 128 blocks (16 output rows each). Block: 96 threads = 3 waves.
// Wave 0 -> i-quarter (q=0), wave 1 -> g-quarter (q=2), wave 2 -> o-quarter (q=3).
// Each wave computes a 16-row strip dot-product via V_WMMA_F32_16X16X4_F32,
// streaming the weight strip with one b64 load per lane per WMMA.
__global__ void lstm_layer_kernel(const float* __restrict__ w_ih,
                                  const float* __restrict__ b_ih,
                                  const float* __restrict__ b_hh,
                                  const float* __restrict__ x,
                                  float* __restrict__ h_out) {
    __shared__ float gsum[3][16];

    const int lane = threadIdx.x & 31;
    const int wv   = threadIdx.x >> 5;           // 0..2
    const int q    = (wv == 0) ? 0 : (wv + 1);   // gate quarter: i=0, g=2, o=3

    const int rowbase = blockIdx.x * 16;         // hidden index of this strip
    const int m       = lane & 15;               // row within strip
    const int kl      = (lane >> 4) << 1;        // K sub-offset: 0 or 2 (A-matrix layout)

    // Per-lane streaming pointers (row stride = HIDDEN floats, 8-byte aligned).
    const float* wa = w_ih + (size_t)(q * HIDDEN + rowbase + m) * HIDDEN + kl;
    const float* xb = x + kl;

    v8f acc = {0.f, 0.f, 0.f, 0.f, 0.f, 0.f, 0.f, 0.f};

    // 2048 / 16 = 128 iterations; 4 WMMAs + 8 independent b64 loads each.
    for (int k = 0; k < HIDDEN; k += 16) {
        v2f a0 = *(const v2f*)(wa + k);
        v2f a1 = *(const v2f*)(wa + k + 4);
        v2f a2 = *(const v2f*)(wa + k + 8);
        v2f a3 = *(const v2f*)(wa + k + 12);
        v2f b0 = *(const v2f*)(xb + k);
        v2f b1 = *(const v2f*)(xb + k + 4);
        v2f b2 = *(const v2f*)(xb + k + 8);
        v2f b3 = *(const v2f*)(xb + k + 12);
        // D = A(16x4) * B(4x16) + C ; B has x broadcast across all 16 columns,
        // so every column of D accumulates the same 16-row GEMV partials.
        acc = __builtin_amdgcn_wmma_f32_16x16x4_f32(false, a0, false, b0,
                                                    (short)0, acc, false, false);
        acc = __builtin_amdgcn_wmma_f32_16x16x4_f32(false, a1, false, b1,
                                                    (short)0, acc, false, false);
        acc = __builtin_amdgcn_wmma_f32_16x16x4_f32(false, a2, false, b2,
                                                    (short)0, acc, false, false);
        acc = __builtin_amdgcn_wmma_f32_16x16x4_f32(false, a3, false, b3,
                                                    (short)0, acc, false, false);
    }

    // C/D layout: VGPR j, lanes 0-15 => M=j, lanes 16-31 => M=8+j.
    // Column 0 (lane 0 / lane 16) carries the strip's 16 dot products.
    if (lane == 0) {
        #pragma unroll
        for (int j = 0; j < 8; ++j) gsum[wv][j] = acc[j];
    } else if (lane == 16) {
        #pragma unroll
        for (int j = 0; j < 8; ++j) gsum[wv][8 + j] = acc[j];
    }
    __syncthreads();

    if (threadIdx.x < 16) {
        const int row = rowbase + (int)threadIdx.x;
        float di = gsum[0][threadIdx.x] + b_ih[row] + b_hh[row];
        float dg = gsum[1][threadIdx.x] + b_ih[2 * HIDDEN + row] + b_hh[2 * HIDDEN + row];
        float dn = gsum[2][threadIdx.x] + b_ih[3 * HIDDEN + row] + b_hh[3 * HIDDEN + row];
        float c  = sigmoidf_(di) * tanhf(dg);       // sigmoid(f)*c_prev == 0
        h_out[row] = sigmoidf_(dn) * tanhf(c);
    }
}

// logits = w_choice @ h + b_choice, prefix-masked by (1 + task).
__global__ void choice_kernel(const float* __restrict__ w_choice,
                              const float* __restrict__ b_choice,
                              const float* __restrict__ h,
                              const int* __restrict__ task,
                              float* __restrict__ out) {
    __shared__ float red[256];
    const int r   = blockIdx.x;          // 0..18
    const int tid = threadIdx.x;

    float s = 0.f;
    const float* wr = w_choice + (size_t)r * HIDDEN;
    for (int k = tid; k < HIDDEN; k += 256) s += wr[k] * h[k];
    red[tid] = s;
    __syncthreads();
    for (int off = 128; off > 0; off >>= 1) {
        if (tid < off) red[tid] += red[tid + off];
        __syncthreads();
    }
    if (tid == 0) {
        float v = red[0] + b_choice[r];
        out[r] = (r < 1 + task[0]) ? v : -1e9f;
    }
}

extern "C" void kernel_launch(void* const* d_in, const int* in_sizes, int n_in,
                              void* d_out, int out_size, void* d_ws, size_t ws_size,
                              hipStream_t stream) {
    const int*   input_idx = (const int*)  d_in[0];
    const int*   task      = (const int*)  d_in[1];
    const float* embedding = (const float*)d_in[2];
    const float* w_ih_0    = (const float*)d_in[3];
    // d_in[4] = w_hh_0 : multiplied by h=0, never read
    const float* b_ih_0    = (const float*)d_in[5];
    const float* b_hh_0    = (const float*)d_in[6];
    const float* w_ih_1    = (const float*)d_in[7];
    // d_in[8] = w_hh_1 : multiplied by h=0, never read
    const float* b_ih_1    = (const float*)d_in[9];
    const float* b_hh_1    = (const float*)d_in[10];
    const float* w_choice  = (const float*)d_in[11];
    const float* b_choice  = (const float*)d_in[12];
    float* out = (float*)d_out;

    float* ws = (float*)d_ws;
    float* x  = ws;               // [HIDDEN]
    float* h0 = ws + HIDDEN;      // [HIDDEN]
    float* h1 = ws + 2 * HIDDEN;  // [HIDDEN]

    embed_gather_kernel<<<HIDDEN / 256, 256, 0, stream>>>(input_idx, embedding, x);
    lstm_layer_kernel<<<HIDDEN / 16, 96, 0, stream>>>(w_ih_0, b_ih_0, b_hh_0, x, h0);
    lstm_layer_kernel<<<HIDDEN / 16, 96, 0, stream>>>(w_ih_1, b_ih_1, b_hh_1, h0, h1);
    choice_kernel<<<CHOICE_NUM, 256, 0, stream>>>(w_choice, b_choice, h1, task, out);
}